// Warp_Object_83305185673884
// MI455X (gfx1250) — compile-verified
//
#include <hip/hip_runtime.h>

// Bicubic warp (Catmull-Rom), B=8, C=3, H=W=1024, fp32.
// Memory-bound gather: roofline floor ~256 MB / 23.3 TB/s ~= 11 us on MI455X.
// Strategy: keep the 96 MB image hot in the 192 MB L2; stream deltas/output
// with non-temporal hints so they don't evict image lines. One thread per
// (b,y,x) pixel handles all 3 channels, amortizing coefficient + index math.

constexpr int B = 8;
constexpr int C = 3;
constexpr int H = 1024;
constexpr int W = 1024;

__device__ __forceinline__ void cubic_coeffs(float t, float c[4]) {
    // Catmull-Rom weights, identical algebra to the reference.
    float t2 = t * t;
    float t3 = t2 * t;
    c[0] = 0.5f * (-t3 + 2.0f * t2 - t);
    c[1] = 0.5f * (3.0f * t3 - 5.0f * t2 + 2.0f);
    c[2] = 0.5f * (-3.0f * t3 + 4.0f * t2 + t);
    c[3] = 1.0f - (c[0] + c[1] + c[2]);
}

__global__ __launch_bounds__(256, 8) void warp_bicubic_kernel(
    const float* __restrict__ img,   // [B,C,H,W]
    const float* __restrict__ dxp,   // [B,H,W]
    const float* __restrict__ dyp,   // [B,H,W]
    float* __restrict__ out)         // [B,C,H,W]
{
    const int p = blockIdx.x * blockDim.x + threadIdx.x;   // pixel id in [0, B*H*W)
    // W,H are powers of two: decompose with shifts/masks.
    const int x = p & (W - 1);
    const int y = (p >> 10) & (H - 1);
    const int b = p >> 20;

    // Streaming, read-once inputs: non-temporal so they don't pollute L2.
    const float fdx = __builtin_nontemporal_load(&dxp[p]);
    const float fdy = __builtin_nontemporal_load(&dyp[p]);

    // Faithful normalize -> un-normalize round trip from the reference
    // (algebraically x+dx, but keep the same rounding sequence).
    const float nx = 2.0f * ((float)x + fdx) / (float)(W - 1) - 1.0f;
    const float ny = 2.0f * ((float)y + fdy) / (float)(H - 1) - 1.0f;
    const float xm = (nx + 1.0f) * ((float)(W - 1) * 0.5f);
    const float ym = (ny + 1.0f) * ((float)(H - 1) * 0.5f);

    const float xf = floorf(xm);
    const float yf = floorf(ym);
    const float tx = xm - xf;
    const float ty = ym - yf;

    float cx[4], cy[4];
    cubic_coeffs(tx, cx);
    cubic_coeffs(ty, cy);

    const int x0 = (int)xf;
    const int y0 = (int)yf;

    int xs[4];   // clamped column indices
    int ys[4];   // clamped row offsets (pre-multiplied by W)
#pragma unroll
    for (int j = 0; j < 4; ++j) {
        xs[j] = min(max(x0 + j - 1, 0), W - 1);
        ys[j] = min(max(y0 + j - 1, 0), H - 1) * W;
    }

    const size_t plane = (size_t)H * W;
    const float* __restrict__ imgb = img + (size_t)b * C * plane;
    const size_t obase = ((size_t)(b * C) * H + y) * W + x;

#pragma unroll
    for (int c = 0; c < C; ++c) {
        const float* __restrict__ chan = imgb + (size_t)c * plane;
        float acc = 0.0f;
#pragma unroll
        for (int i = 0; i < 4; ++i) {
            const float* __restrict__ row = chan + ys[i];
            // 4-tap horizontal pass (gathers served mostly from L2: image fits
            // in the 192 MB L2 and neighboring pixels share taps ~16x).
            float r = row[xs[0]] * cx[0];
            r = fmaf(row[xs[1]], cx[1], r);
            r = fmaf(row[xs[2]], cx[2], r);
            r = fmaf(row[xs[3]], cx[3], r);
            acc = fmaf(r, cy[i], acc);
        }
        // Write-once output: non-temporal store, keep L2 for the image.
        __builtin_nontemporal_store(acc, &out[obase + c * plane]);
    }
}

extern "C" void kernel_launch(void* const* d_in, const int* in_sizes, int n_in,
                              void* d_out, int out_size, void* d_ws, size_t ws_size,
                              hipStream_t stream) {
    const float* img = (const float*)d_in[0];   // [8,3,1024,1024] fp32
    const float* dx  = (const float*)d_in[1];   // [8,1024,1024]   fp32
    const float* dy  = (const float*)d_in[2];   // [8,1024,1024]   fp32
    float* out = (float*)d_out;                 // [8,3,1024,1024] fp32

    const int total = B * H * W;                // 2^23 pixels, one thread each
    const int block = 256;                      // 8 wave32 waves per workgroup
    const int grid  = total / block;            // exact cover (2^23 / 256)

    warp_bicubic_kernel<<<grid, block, 0, stream>>>(img, dx, dy, out);
}